// DihedralGroupConv_42245298323946
// MI455X (gfx1250) — compile-verified
//
#include <hip/hip_runtime.h>

typedef __attribute__((ext_vector_type(2))) float v2f;
typedef __attribute__((ext_vector_type(4))) float v4f;
typedef __attribute__((ext_vector_type(8))) float v8f;

#define NGEN   4
#define TWO_N  400
#define NCH    32
#define NB     512
#define OTILES (TWO_N / 16)   // 25 o-tiles of 16

// ---------------------------------------------------------------------------
// Kernel 1: invert the dense permutation matrices.
// pinv[g][o] = i such that perm[g][i][o] == 1   (so x_g[.,o] = x[., pinv[g][o]])
// 1600 threads, coalesced across o; perm (2.56 MB) is L2-resident.
// ---------------------------------------------------------------------------
__global__ void build_invperm_kernel(const float* __restrict__ perm,
                                     int* __restrict__ pinv) {
    int t = blockIdx.x * blockDim.x + threadIdx.x;
    if (t >= NGEN * TWO_N) return;
    int g = t / TWO_N;
    int o = t - g * TWO_N;
    const float* p = perm + (size_t)g * TWO_N * TWO_N + o;
    int found = 0;
    for (int i = 0; i < TWO_N; ++i) {
        if (p[(size_t)i * TWO_N] > 0.5f) found = i;
    }
    pinv[t] = found;
}

// ---------------------------------------------------------------------------
// Kernel 2: out[b,d,o] = sum_g sum_c weight[g,c,d] * x[b,c,pinv[g][o]]
// One wave per (b, 16-wide o-tile). Output tile: M=16 (o) x N=32 (d),
// accumulated over K=32 channels x 4 generators with v_wmma_f32_16x16x4_f32.
// ---------------------------------------------------------------------------
__global__ void __launch_bounds__(256)
dihedral_gconv_wmma(const float* __restrict__ x,
                    const float* __restrict__ w,
                    const int*  __restrict__ pinv,
                    float* __restrict__ out) {
    const int lane  = threadIdx.x & 31;
    const int wid   = threadIdx.x >> 5;
    const int wave  = blockIdx.x * 8 + wid;       // 0 .. 12799 (exact)
    const int b     = wave / OTILES;
    const int ot    = wave - b * OTILES;
    const int o0    = ot * 16;
    const int m     = lane & 15;                  // row-within-tile / N index
    const int khalf = lane >> 4;                  // 0: K=0,1  1: K=2,3

    v8f acc0 = {};   // N-tile d = 0..15
    v8f acc1 = {};   // N-tile d = 16..31

    const float* xb = x + (size_t)b * (NCH * TWO_N);

    for (int g = 0; g < NGEN; ++g) {
        // source column in x for this lane's output row o0+m
        const int idx = pinv[g * TWO_N + o0 + m];
        const float* xcol = xb + idx;
        const float* wg   = w + g * (NCH * NCH);  // [c][d], row stride 32

#pragma unroll
        for (int kb = 0; kb < 8; ++kb) {
            const int c0 = kb * 4 + khalf * 2;    // this lane's two K (=channel) values

            // A fragment: A[m, k] = x[b, c0..c0+1, idx]   (16x4, f32, 2 VGPRs)
            v2f a;
            a.x = xcol[(c0 + 0) * TWO_N];
            a.y = xcol[(c0 + 1) * TWO_N];

            // B fragments: B[k, n] = weight[g, c0..c0+1, n(+16)]  (4x16 each)
            v2f b0, b1;
            b0.x = wg[(c0 + 0) * NCH + m];
            b0.y = wg[(c0 + 1) * NCH + m];
            b1.x = wg[(c0 + 0) * NCH + 16 + m];
            b1.y = wg[(c0 + 1) * NCH + 16 + m];

            acc0 = __builtin_amdgcn_wmma_f32_16x16x4_f32(
                       false, a, false, b0, (short)0, acc0, false, false);
            acc1 = __builtin_amdgcn_wmma_f32_16x16x4_f32(
                       false, a, false, b1, (short)0, acc1, false, false);
        }
    }

    // C/D layout: lane holds D[M = v + 8*khalf, N = m] in acc[v], v = 0..7.
    // M -> o = o0 + 8*khalf + v  (contiguous in v)  ;  N -> d.
    // => each lane stores 8 contiguous floats per N-tile: two b128 stores.
    float* op0 = out + (size_t)b * (NCH * TWO_N) + (size_t)m * TWO_N
                     + o0 + khalf * 8;            // d = m
    v4f s;
    s = (v4f){acc0[0], acc0[1], acc0[2], acc0[3]};
    *reinterpret_cast<v4f*>(op0) = s;
    s = (v4f){acc0[4], acc0[5], acc0[6], acc0[7]};
    *reinterpret_cast<v4f*>(op0 + 4) = s;

    float* op1 = op0 + 16 * TWO_N;                // d = m + 16
    s = (v4f){acc1[0], acc1[1], acc1[2], acc1[3]};
    *reinterpret_cast<v4f*>(op1) = s;
    s = (v4f){acc1[4], acc1[5], acc1[6], acc1[7]};
    *reinterpret_cast<v4f*>(op1 + 4) = s;
}

// ---------------------------------------------------------------------------
extern "C" void kernel_launch(void* const* d_in, const int* in_sizes, int n_in,
                              void* d_out, int out_size, void* d_ws, size_t ws_size,
                              hipStream_t stream) {
    const float* x    = (const float*)d_in[0];   // [512, 32, 400] f32
    const float* w    = (const float*)d_in[1];   // [4, 32, 32]   f32
    const float* perm = (const float*)d_in[2];   // [4, 400, 400] f32
    float* out = (float*)d_out;                  // [512, 32, 400] f32
    int* pinv = (int*)d_ws;                      // 4*400 ints = 6.4 KB scratch

    build_invperm_kernel<<<(NGEN * TWO_N + 255) / 256, 256, 0, stream>>>(perm, pinv);

    const int total_waves = NB * OTILES;         // 12800
    dihedral_gconv_wmma<<<total_waves / 8, 256, 0, stream>>>(x, w, pinv, out);
}